// QPU_10574209482758
// MI455X (gfx1250) — compile-verified
//
#include <hip/hip_runtime.h>
#include <stdint.h>

typedef unsigned int u32x4 __attribute__((ext_vector_type(4)));
typedef int          i32x4 __attribute__((ext_vector_type(4)));
typedef int          i32x8 __attribute__((ext_vector_type(8)));

#define C_IN       64
#define C_OUT      128
#define WAVES      8
#define WT_STRIDE  132   // padded row stride (floats): (132c+4l)%64 spreads banks

__global__ __launch_bounds__(256)
void qpu_kernel(const float* __restrict__ x,
                const float* __restrict__ weight,
                const float* __restrict__ bias,
                float* __restrict__ out,
                int tokens)
{
    __shared__ __align__(16) float  rawW[C_OUT * C_IN];     // 32 KB, TDM destination
    __shared__ __align__(16) float  wT[C_IN * WT_STRIDE];   // transposed [c][o], padded
    __shared__ __align__(16) float4 tokD[WAVES][C_IN];      // per wave: {theta, i*, j*, k*}

    const int tid  = threadIdx.x;
    const int wave = tid >> 5;
    const int lane = tid & 31;

    // ---- Stage 1: TDM async DMA of weight (128x64 f32, contiguous 32 KB) -> LDS ----
#if defined(__HIP_DEVICE_COMPILE__)
    if (wave == 0) {
        const uint64_t gaddr = (uint64_t)(uintptr_t)weight;
        const uint32_t laddr = (uint32_t)(uintptr_t)&rawW[0];   // low 32 bits = LDS offset
        const uint32_t N     = (uint32_t)(C_OUT * C_IN);        // 8192 elements

        u32x4 g0;
        g0[0] = 1u;                                        // count=1, user descriptor
        g0[1] = laddr;                                     // lds_addr (bytes)
        g0[2] = (uint32_t)(gaddr & 0xFFFFFFFFu);           // global_addr[31:0]
        g0[3] = (uint32_t)((gaddr >> 32) & 0x01FFFFFFu)    // global_addr[56:32]
              | (2u << 30);                                // type = 2 ("image")

        i32x8 g1;
        g1[0] = (int)(2u << 16);                           // data_size=4B, wg_mask=0, no pad/iter
        g1[1] = (int)((N & 0xFFFFu) << 16);                // tensor_dim0[15:0] @ bits 63:48
        g1[2] = (int)((1u << 16) | (N >> 16));             // tensor_dim1=1, tensor_dim0[31:16]
        g1[3] = (int)((N & 0xFFFFu) << 16);                // tile_dim0 @ bits 127:112
        g1[4] = 1;                                         // tile_dim1=1, tile_dim2=0
        g1[5] = (int)N;                                    // tensor_dim0_stride[31:0]
        g1[6] = 0;                                         // stride hi, dim1_stride lo
        g1[7] = 0;

        i32x4 gz = {0, 0, 0, 0};                           // groups 2/3 unused (<=2D)
#if __has_builtin(__builtin_amdgcn_tensor_load_to_lds)
#if __clang_major__ >= 23
        i32x8 gz8 = {0, 0, 0, 0, 0, 0, 0, 0};
        __builtin_amdgcn_tensor_load_to_lds(g0, g1, gz, gz, gz8, 0);
#else
        __builtin_amdgcn_tensor_load_to_lds(g0, g1, gz, gz, 0);
#endif
        __builtin_amdgcn_s_wait_tensorcnt(0);
#else
        // Fallback: cooperative vector loads
        for (int idx = lane * 4; idx < C_OUT * C_IN; idx += 32 * 4)
            *(float4*)&rawW[idx] = *(const float4*)&weight[idx];
#endif
    }
#endif
    __syncthreads();

    // ---- Stage 2: transpose weight into wT[c][o] (one-time, padded stride) ----
    for (int idx = tid; idx < C_OUT * C_IN; idx += 256) {
        int o = idx >> 6;          // idx / C_IN
        int c = idx & (C_IN - 1);  // idx % C_IN
        wT[c * WT_STRIDE + o] = rawW[idx];
    }
    __syncthreads();

    // ---- Stage 3: per-token precompute: theta_base, unit axis (i,j,k)/|v| ----
    const int  token = blockIdx.x * WAVES + wave;
    const bool valid = token < tokens;
    if (valid) {
        const float* xr = x + (size_t)token * (4 * C_IN);
#pragma unroll
        for (int h = 0; h < 2; ++h) {
            int   c = lane + 32 * h;
            float r = xr[c];
            float i = xr[C_IN + c];
            float j = xr[2 * C_IN + c];
            float k = xr[3 * C_IN + c];
            float rc  = fminf(fmaxf(r, -0.999999f), 0.999999f);   // CLAMP = 1 - 1e-6
            float th  = acosf(rc);
            float inv = 1.0f / sqrtf(fmaf(i, i, fmaf(j, j, fmaf(k, k, 1e-12f))));
            tokD[wave][c] = make_float4(th, i * inv, j * inv, k * inv);
        }
    }
    __syncthreads();

    // ---- Stage 4: 4 Hamilton chains per lane (o = 4*lane + u) ----
    if (valid) {
        const float4 bb = reinterpret_cast<const float4*>(bias)[lane];
        const float  bv[4] = {bb.x, bb.y, bb.z, bb.w};
        float ar[4] = {1.f, 1.f, 1.f, 1.f};
        float ai[4] = {0.f, 0.f, 0.f, 0.f};
        float aj[4] = {0.f, 0.f, 0.f, 0.f};
        float ak[4] = {0.f, 0.f, 0.f, 0.f};

#pragma unroll 2
        for (int c = 0; c < C_IN; ++c) {
            const float4 t  = tokD[wave][c];                                        // broadcast
            const float4 wv = *reinterpret_cast<const float4*>(&wT[c * WT_STRIDE + 4 * lane]);
            const float  wvu[4] = {wv.x, wv.y, wv.z, wv.w};
#pragma unroll
            for (int u = 0; u < 4; ++u) {
                float th = fmaf(wvu[u], t.x, bv[u]);
                float s, cs;
                __sincosf(th, &s, &cs);                 // v_sin_f32 / v_cos_f32
                float qr = cs, qi = t.y * s, qj = t.z * s, qk = t.w * s;
                // acc = acc (x) q   (left-to-right Hamilton product)
                float nr = ar[u]*qr; nr = fmaf(-ai[u],qi,nr); nr = fmaf(-aj[u],qj,nr); nr = fmaf(-ak[u],qk,nr);
                float ni = ar[u]*qi; ni = fmaf( ai[u],qr,ni); ni = fmaf( aj[u],qk,ni); ni = fmaf(-ak[u],qj,ni);
                float nj = ar[u]*qj; nj = fmaf(-ai[u],qk,nj); nj = fmaf( aj[u],qr,nj); nj = fmaf( ak[u],qi,nj);
                float nk = ar[u]*qk; nk = fmaf( ai[u],qj,nk); nk = fmaf(-aj[u],qi,nk); nk = fmaf( ak[u],qr,nk);
                ar[u] = nr; ai[u] = ni; aj[u] = nj; ak[u] = nk;
            }
        }

        // ---- Stage 5: normalize and store (coalesced float4 per component) ----
        float outr[4], outi[4], outj[4], outk[4];
#pragma unroll
        for (int u = 0; u < 4; ++u) {
            float invn = 1.0f / sqrtf(fmaf(ar[u], ar[u],
                              fmaf(ai[u], ai[u],
                              fmaf(aj[u], aj[u],
                              fmaf(ak[u], ak[u], 1e-12f)))));
            outr[u] = ar[u] * invn;
            outi[u] = ai[u] * invn;
            outj[u] = aj[u] * invn;
            outk[u] = ak[u] * invn;
        }
        float* op = out + (size_t)token * (4 * C_OUT) + 4 * lane;
        *reinterpret_cast<float4*>(op)             = make_float4(outr[0], outr[1], outr[2], outr[3]);
        *reinterpret_cast<float4*>(op + C_OUT)     = make_float4(outi[0], outi[1], outi[2], outi[3]);
        *reinterpret_cast<float4*>(op + 2 * C_OUT) = make_float4(outj[0], outj[1], outj[2], outj[3]);
        *reinterpret_cast<float4*>(op + 3 * C_OUT) = make_float4(outk[0], outk[1], outk[2], outk[3]);
    }
}

extern "C" void kernel_launch(void* const* d_in, const int* in_sizes, int n_in,
                              void* d_out, int out_size, void* d_ws, size_t ws_size,
                              hipStream_t stream)
{
    (void)n_in; (void)out_size; (void)d_ws; (void)ws_size;
    const float* x      = (const float*)d_in[0];
    const float* weight = (const float*)d_in[1];
    const float* bias   = (const float*)d_in[2];
    float*       out    = (float*)d_out;

    const int tokens = in_sizes[0] / (4 * C_IN);          // B*T = 2048
    const int blocks = (tokens + WAVES - 1) / WAVES;      // one token per wave
    qpu_kernel<<<blocks, 256, 0, stream>>>(x, weight, bias, out, tokens);
}